// DotProductIncident_26207890440258
// MI455X (gfx1250) — compile-verified
//
#include <hip/hip_runtime.h>

// D = A(16x4) * B(4x16) + C, f32. B = all-ones turns the WMMA into a
// cross-lane row-sum engine: D[m][n] = sum_k A[m][k].
typedef __attribute__((ext_vector_type(2))) float v2f;
typedef __attribute__((ext_vector_type(8))) float v8f;

#define D_FEAT 64
#define WAVES_PER_BLOCK 8
#define EDGES_PER_WAVE 16
#define EDGES_PER_BLOCK (WAVES_PER_BLOCK * EDGES_PER_WAVE)

__global__ __launch_bounds__(256) void edge_dot_wmma_kernel(
    const float* __restrict__ nf,    // [N_NODES, 64]
    const int*   __restrict__ esrc,  // [E]
    const int*   __restrict__ edst,  // [E]
    float*       __restrict__ out,   // [E]
    int E)
{
    const int lane = threadIdx.x & 31;
    const int wave = threadIdx.x >> 5;
    const int base = (blockIdx.x * WAVES_PER_BLOCK + wave) * EDGES_PER_WAVE;
    if (base >= E) return;                 // wave-uniform exit: EXEC stays all-1s

    const int m = lane & 15;               // edge row within the 16-edge tile (A-matrix M)
    const int h = lane >> 4;               // K-half: lanes 0-15 -> K={0,1}, 16-31 -> K={2,3}

    int e = base + m;
    if (e >= E) e = E - 1;                 // clamp tail lanes (loads only; stores are guarded)
    const int si = esrc[e];
    const int di = edst[e];

    // 32-bit element offsets (max 100000*64 = 6.4M, fits easily) so the
    // compiler can use saddr(nf) + 32-bit voffset addressing on the hot loads.
    const int soff = si * D_FEAT + h * 32; // lane owns contiguous 128B half-row
    const int doff = di * D_FEAT + h * 32;

    // Partial dot products feeding the 16x4 f32 A-matrix layout:
    //   VGPR0 (a0) = K = 2h   : features [32h,      32h + 16)
    //   VGPR1 (a1) = K = 2h+1 : features [32h + 16, 32h + 32)
    float a0 = 0.0f, a1 = 0.0f;
#pragma unroll
    for (int j = 0; j < 4; ++j) {
        float4 s = *(const float4*)(nf + soff + 4 * j);
        float4 d = *(const float4*)(nf + doff + 4 * j);
        a0 = fmaf(s.x, d.x, a0); a0 = fmaf(s.y, d.y, a0);
        a0 = fmaf(s.z, d.z, a0); a0 = fmaf(s.w, d.w, a0);
    }
#pragma unroll
    for (int j = 4; j < 8; ++j) {
        float4 s = *(const float4*)(nf + soff + 4 * j);
        float4 d = *(const float4*)(nf + doff + 4 * j);
        a1 = fmaf(s.x, d.x, a1); a1 = fmaf(s.y, d.y, a1);
        a1 = fmaf(s.z, d.z, a1); a1 = fmaf(s.w, d.w, a1);
    }

    v2f A = {a0, a1};
    v2f B = {1.0f, 1.0f};                  // ones -> WMMA computes row sums of A
    v8f C = {};
    // 8 args: (neg_a, A, neg_b, B, c_mod, C, reuse_a, reuse_b)
    v8f D = __builtin_amdgcn_wmma_f32_16x16x4_f32(
        /*neg_a=*/false, A, /*neg_b=*/false, B,
        /*c_mod=*/(short)0, C, /*reuse_a=*/false, /*reuse_b=*/false);

    // D layout: VGPR r, lanes 0-15 hold (M=r, N=lane); lanes 16-31 hold (M=8+r).
    // Column N=0 lives in lanes 0 and 16 -> each holds 8 edge scores in D[0..7].
    if (m == 0) {
        const int off = base + h * 8;
        if (off + 8 <= E) {                // hot path: this lane's 8 edges all valid
            float4 o0 = make_float4(D[0], D[1], D[2], D[3]);
            float4 o1 = make_float4(D[4], D[5], D[6], D[7]);
            *(float4*)(out + off)     = o0;
            *(float4*)(out + off + 4) = o1;
        } else {                           // cold tail: compact guarded loop
            const int n = E - off;         // n < 8 here; may be <= 0 for h = 1
            for (int j = 0; j < n; ++j)
                out[off + j] = D[j];
        }
    }
}

extern "C" void kernel_launch(void* const* d_in, const int* in_sizes, int n_in,
                              void* d_out, int out_size, void* d_ws, size_t ws_size,
                              hipStream_t stream) {
    const float* nf   = (const float*)d_in[0];   // node_feature [N, 64] f32
    const int*   esrc = (const int*)  d_in[1];   // edge_src [E] i32
    const int*   edst = (const int*)  d_in[2];   // edge_dst [E] i32
    float*       out  = (float*)d_out;           // [E, 1] f32

    const int E = in_sizes[1];
    const int grid = (E + EDGES_PER_BLOCK - 1) / EDGES_PER_BLOCK;
    edge_dot_wmma_kernel<<<grid, 256, 0, stream>>>(nf, esrc, edst, out, E);
}